// MultVariateKLD_24172075941906
// MI455X (gfx1250) — compile-verified
//
#include <hip/hip_runtime.h>

// KL( N(mu1, diag(v1)) || N(mu2, diag(v2)) ), mean over batch.
// Flat reduction: S = sum over all elements of
//     log(v2) - log(v1) + v1/v2 + (mu2-mu1)^2 / v2
// out = 0.5 * (S - B*N) / B, with B*N = total elements, B = 16384 (reference).
//
// Bandwidth-bound: 268 MB @ 23.3 TB/s ~ 11.5 us floor. Per-element math uses
// raw v_log_f32 / v_rcp_f32 (inputs are in [0.1,1.1]: normal range, ~1 ulp),
// keeping VALU well under the memory floor. The reduction runs through the
// full-precision fp32 WMMA (v_wmma_f32_16x16x4_f32) with B == ones, so the
// matrix pipe acts as a 64-slot exact accumulator per wave. Loop control is
// scalar (uniform trip count) so EXEC provably stays all-ones at every WMMA.

typedef float v2f __attribute__((ext_vector_type(2)));
typedef float v8f __attribute__((ext_vector_type(8)));

#define LN2F 0.69314718055994530942f

__device__ __forceinline__ float kl_term(float m1, float m2, float s1, float s2) {
    float r = __builtin_amdgcn_rcpf(s2);                     // v_rcp_f32 (~1 ulp)
    float d = m2 - m1;
    float lg = __builtin_amdgcn_logf(s2) - __builtin_amdgcn_logf(s1); // log2 diff
    return fmaf(lg, LN2F, (fmaf(d, d, s1)) * r);             // ln-diff + (v1+d^2)/v2
}

__global__ __launch_bounds__(256) void kl_partial_kernel(
    const float* __restrict__ mu1, const float* __restrict__ mu2,
    const float* __restrict__ va1, const float* __restrict__ va2,
    float* __restrict__ partial, unsigned long long nTotal)
{
    const unsigned tid   = threadIdx.x;
    const unsigned gsize = gridDim.x * blockDim.x;
    const unsigned gid   = blockIdx.x * blockDim.x + tid;

    const unsigned long long n4 = nTotal >> 2;          // float4 groups
    const unsigned nIter = (unsigned)(n4 / gsize);      // SCALAR (uniform) trip count
    const unsigned long long nUniform = (unsigned long long)nIter * gsize;

    const float4* p1 = reinterpret_cast<const float4*>(mu1) + gid;
    const float4* p2 = reinterpret_cast<const float4*>(mu2) + gid;
    const float4* p3 = reinterpret_cast<const float4*>(va1) + gid;
    const float4* p4 = reinterpret_cast<const float4*>(va2) + gid;

    v8f acc = {};                   // WMMA C/D accumulator (16x16 f32, 8 regs/lane)
    v2f ones; ones.x = 1.0f; ones.y = 1.0f;   // B matrix: all ones (layout-agnostic)

    // Scalar loop bound -> s_cbranch control, EXEC all-ones at every WMMA.
    #pragma unroll 2
    for (unsigned it = 0; it < nIter; ++it) {
        float4 a = *p1;  p1 += gsize;
        float4 b = *p2;  p2 += gsize;
        float4 p = *p3;  p3 += gsize;
        float4 q = *p4;  p4 += gsize;

        float t0 = kl_term(a.x, b.x, p.x, q.x);
        float t1 = kl_term(a.y, b.y, p.y, q.y);
        float t2 = kl_term(a.z, b.z, p.z, q.z);
        float t3 = kl_term(a.w, b.w, p.w, q.w);

        v2f A; A.x = t0 + t1; A.y = t2 + t3;   // 2 A slots per lane (16x4 f32 A)
        acc = __builtin_amdgcn_wmma_f32_16x16x4_f32(
            /*neg_a=*/false, A, /*neg_b=*/false, ones,
            /*c_mod=*/(short)0, acc, /*reuse_a=*/false, /*reuse_b=*/false);
    }

    // D rows replicated across columns: lane sum of 8 regs == sum of rows 0..7
    // (lanes 0-15) or rows 8..15 (lanes 16-31). One xor-16 gives the wave total.
    float s = acc[0] + acc[1] + acc[2] + acc[3] + acc[4] + acc[5] + acc[6] + acc[7];
    s += __shfl_xor(s, 16);

    // Scalar tail (empty when nTotal divisible by 4*gsize, which it is here).
    float extra = 0.0f;
    for (unsigned long long e = (nUniform << 2) + gid; e < nTotal; e += gsize)
        extra += kl_term(mu1[e], mu2[e], va1[e], va2[e]);
    for (int off = 1; off < 32; off <<= 1)
        extra += __shfl_xor(extra, off);
    s += extra;

    // Cross-wave combine within the block (8 waves), fixed order -> deterministic.
    __shared__ float waveSum[8];
    if ((tid & 31u) == 0u) waveSum[tid >> 5] = s;
    __syncthreads();
    if (tid == 0) {
        float bs = 0.0f;
        #pragma unroll
        for (int w = 0; w < 8; ++w) bs += waveSum[w];
        partial[blockIdx.x] = bs;
    }
}

__global__ __launch_bounds__(1024) void kl_final_kernel(
    const float* __restrict__ partial, int nPartial,
    float* __restrict__ out, float invB, float bnTerm)
{
    __shared__ float sh[1024];
    int t = threadIdx.x;
    sh[t] = (t < nPartial) ? partial[t] : 0.0f;
    __syncthreads();
    #pragma unroll
    for (int s = 512; s > 0; s >>= 1) {
        if (t < s) sh[t] += sh[t + s];
        __syncthreads();
    }
    if (t == 0) out[0] = 0.5f * (sh[0] - bnTerm) * invB;
}

extern "C" void kernel_launch(void* const* d_in, const int* in_sizes, int n_in,
                              void* d_out, int out_size, void* d_ws, size_t ws_size,
                              hipStream_t stream) {
    const float* mu1 = (const float*)d_in[0];
    const float* mu2 = (const float*)d_in[1];
    const float* va1 = (const float*)d_in[2];
    const float* va2 = (const float*)d_in[3];
    float* out = (float*)d_out;
    float* ws  = (float*)d_ws;   // 1024 floats of block partials

    const unsigned long long total = (unsigned long long)in_sizes[0]; // B*N = 16777216
    const int BLOCKS = 1024, THREADS = 256;   // 2^18 threads -> 16 uniform f4 iters

    // Reference fixes B = 16384, N = 1024; sum of per-row (-n) is -(B*N) = -total.
    const float Bf = 16384.0f;

    kl_partial_kernel<<<BLOCKS, THREADS, 0, stream>>>(mu1, mu2, va1, va2, ws, total);
    kl_final_kernel<<<1, 1024, 0, stream>>>(ws, BLOCKS, out, 1.0f / Bf, (float)total);
}